// GNActGCTDWConv2d_33354716021306
// MI455X (gfx1250) — compile-verified
//
#include <hip/hip_runtime.h>
#include <hip/hip_bf16.h>
#include <stdint.h>

// Problem constants (fixed by setup_inputs)
#define HH 128
#define WW 128
#define HWPIX 16384
#define BS 8
#define CC 256
#define GROUPS 32
#define NPG (8 * HWPIX)   // elements per (batch, group) = 131072
#define PCHUNKS 64
#define PPB 256           // pixels handled per block chunk
#define EPSF 1e-5f

typedef unsigned int u32x4 __attribute__((ext_vector_type(4)));
typedef int i32x8 __attribute__((ext_vector_type(8)));
typedef int i32x4 __attribute__((ext_vector_type(4)));

// ---------------------------------------------------------------------------
// Pass 1a: per-(b, p-chunk, group) partial sum / sumsq.
// Layout of x: [(p*BS + b)*CC + c]; for fixed p one block reads a contiguous
// 1KB channel vector (perfect wave32 coalescing).
// ---------------------------------------------------------------------------
__global__ __launch_bounds__(256) void k_stats_partial(
    const float* __restrict__ x, float* __restrict__ part1) {
  const int pc = blockIdx.x;
  const int b = blockIdx.y;
  const int ch = threadIdx.x;
  const float* px = x + ((size_t)(pc * PPB) * BS + b) * CC + ch;
  float s = 0.f, s2 = 0.f;
  for (int i = 0; i < PPB; ++i) {
    float v = px[(size_t)i * BS * CC];
    s += v;
    s2 += v * v;
  }
  __shared__ float rs[256];
  __shared__ float rq[256];
  rs[ch] = s;
  rq[ch] = s2;
  __syncthreads();
  if (ch < 32) {  // one thread per group, sum its 8 channels
    float a = 0.f, q = 0.f;
#pragma unroll
    for (int j = 0; j < 8; ++j) {
      a += rs[ch * 8 + j];
      q += rq[ch * 8 + j];
    }
    float* o = part1 + ((size_t)(b * PCHUNKS + pc) * 32 + ch) * 2;
    o[0] = a;
    o[1] = q;
  }
}

// ---------------------------------------------------------------------------
// Pass 1b: finalize mean / rstd per (b, group). Fixed-order sum -> deterministic.
// ---------------------------------------------------------------------------
__global__ __launch_bounds__(256) void k_stats_final(
    const float* __restrict__ part1, float* __restrict__ meanrstd) {
  const int t = threadIdx.x;  // t = b*32 + g
  const int b = t >> 5;
  const int g = t & 31;
  float s = 0.f, s2 = 0.f;
  for (int pc = 0; pc < PCHUNKS; ++pc) {
    const float* p = part1 + ((size_t)(b * PCHUNKS + pc) * 32 + g) * 2;
    s += p[0];
    s2 += p[1];
  }
  const float inv = 1.0f / (float)NPG;
  float mean = s * inv;
  float var = s2 * inv - mean * mean;
  var = fmaxf(var, 0.0f);
  meanrstd[t * 2 + 0] = mean;
  meanrstd[t * 2 + 1] = rsqrtf(var + EPSF);
}

// ---------------------------------------------------------------------------
// Pass 2: xa = GELU(GN_affine(x)); store xa; partial sum of xa^2 per (b,c).
// ---------------------------------------------------------------------------
__global__ __launch_bounds__(256) void k_norm_gelu(
    const float* __restrict__ x, const float* __restrict__ gn_w,
    const float* __restrict__ gn_b, const float* __restrict__ meanrstd,
    float* __restrict__ xa, float* __restrict__ part2) {
  const int pc = blockIdx.x;
  const int b = blockIdx.y;
  const int ch = threadIdx.x;
  const int g = ch >> 3;
  const float mean = meanrstd[(b * 32 + g) * 2 + 0];
  const float rstd = meanrstd[(b * 32 + g) * 2 + 1];
  const float sw = gn_w[ch] * rstd;
  const float sb = gn_b[ch] - mean * sw;  // xn = x*sw + sb
  const float kInvSqrt2 = 0.70710678118654752440f;
  const size_t base = ((size_t)(pc * PPB) * BS + b) * CC + ch;
  float acc = 0.f;
  for (int i = 0; i < PPB; ++i) {
    size_t idx = base + (size_t)i * BS * CC;
    float xn = x[idx] * sw + sb;
    float a = 0.5f * xn * (1.0f + erff(xn * kInvSqrt2));  // exact GELU
    xa[idx] = a;
    acc += a * a;
  }
  part2[(size_t)(b * PCHUNKS + pc) * CC + ch] = acc;
}

// ---------------------------------------------------------------------------
// Pass 3: GCT gate per (b,c): emb = sqrt(sum xa^2 + eps)*alpha;
// nrm = gamma * rsqrt(mean_c(emb^2) + eps); gate = 1 + tanh(emb*nrm + beta).
// ---------------------------------------------------------------------------
__global__ __launch_bounds__(256) void k_gate(
    const float* __restrict__ part2, const float* __restrict__ alpha,
    const float* __restrict__ gamma, const float* __restrict__ beta,
    float* __restrict__ gate) {
  const int b = blockIdx.x;
  const int c = threadIdx.x;
  float s2 = 0.f;
  for (int pc = 0; pc < PCHUNKS; ++pc)
    s2 += part2[(size_t)(b * PCHUNKS + pc) * CC + c];
  float emb = sqrtf(s2 + EPSF) * alpha[c];
  __shared__ float red[256];
  red[c] = emb * emb;
  __syncthreads();
  for (int off = 128; off > 0; off >>= 1) {
    if (c < off) red[c] += red[c + off];
    __syncthreads();
  }
  float m = red[0] * (1.0f / 256.0f);
  float nrm = gamma[c] * rsqrtf(m + EPSF);
  gate[b * CC + c] = 1.0f + tanhf(emb * nrm + beta[c]);
}

// ---------------------------------------------------------------------------
// Pass 4: depthwise 5x5 conv via Tensor Data Mover.
// Block = (32 channels, 16x16 pixel tile, one batch). The halo slab
// (<=20 x <=20 pixels x 32 channels) is a 3D strided tensor tile:
//   dim0 = channels (contiguous, 128B), dim1 = x (stride 8KB), dim2 = y (1MB).
// One TENSOR_LOAD_TO_LDS moves the whole tile into packed LDS.
// Borders are handled by clipping the tile dims (uniform per block) and
// zero-filling clipped taps in the compute loop. gate folds into the epilogue.
// ---------------------------------------------------------------------------
#define TILE 16
#define HALO 20

__global__ __launch_bounds__(256) void k_dwconv(
    const float* __restrict__ xa, const float* __restrict__ conv_w,
    const float* __restrict__ gate, float* __restrict__ out) {
  __shared__ float tile[HALO * HALO * 32];  // packed [ny][nx][32], <=50KB

  const int cb = blockIdx.x;               // channel block (8 of them)
  const int gy = blockIdx.y >> 3;          // 8x8 tiles of 16
  const int gx = blockIdx.y & 7;
  const int b = blockIdx.z;
  const int c0 = cb * 32;

  const int y0 = gy * TILE - 2;
  const int x0 = gx * TILE - 2;
  const int yA = (y0 < 0) ? 0 : y0;
  const int xA = (x0 < 0) ? 0 : x0;
  const int yB = (y0 + HALO > HH) ? HH : (y0 + HALO);
  const int xB = (x0 + HALO > WW) ? WW : (x0 + HALO);
  const int ny = yB - yA, nx = xB - xA;    // valid tile extents (18 or 20)
  const int oy = yA - y0, ox = xA - x0;    // halo-space origin of valid data

  if (threadIdx.x < 32) {  // wave 0 issues the tensor DMA
    unsigned lds_off = (unsigned)(uintptr_t)(&tile[0]);
    unsigned long long gaddr =
        (unsigned long long)(uintptr_t)(xa + ((size_t)(yA * WW + xA) * BS + b) * CC + c0);
    const unsigned long long str0 = (unsigned long long)(BS * CC);       // x stride (elems)
    const unsigned long long str1 = (unsigned long long)(WW * BS * CC);  // y stride (elems)
    const unsigned td0 = 32, td1 = (unsigned)nx, td2 = (unsigned)ny;

    u32x4 g0 = {};
    g0[0] = 1u;                                        // count=1, user descriptor
    g0[1] = lds_off;                                   // lds_addr (bytes)
    g0[2] = (unsigned)(gaddr & 0xFFFFFFFFull);         // global_addr[31:0]
    g0[3] = (unsigned)((gaddr >> 32) & 0x1FFFFFFull)   // global_addr[56:32]
            | 0x80000000u;                             // type=2 ("image")

    i32x8 g1 = {};
    g1[0] = (int)(2u << 16);                           // data_size = 4 bytes
    g1[1] = (int)((td0 & 0xFFFFu) << 16);              // tensor_dim0[15:0]
    g1[2] = (int)(((td0 >> 16) & 0xFFFFu) | ((td1 & 0xFFFFu) << 16));
    g1[3] = (int)(((td1 >> 16) & 0xFFFFu) | (32u << 16));          // tile_dim0 = 32
    g1[4] = (int)(((unsigned)nx & 0xFFFFu) | (((unsigned)ny & 0xFFFFu) << 16));
    g1[5] = (int)(unsigned)(str0 & 0xFFFFFFFFull);                 // dim0 stride lo
    g1[6] = (int)((unsigned)((str0 >> 32) & 0xFFFFull) |
                  (unsigned)((str1 & 0xFFFFull) << 16));           // s0 hi | s1 lo
    g1[7] = (int)(unsigned)((str1 >> 16) & 0xFFFFFFFFull);         // dim1 stride hi

    i32x4 g2 = {};
    g2[0] = (int)td2;        // tensor_dim2 = ny
    g2[1] = 1;               // tensor_dim3 (unused)
    g2[2] = 0;               // tensor_dim2_stride lo (unused)
    g2[3] = 0;               // stride hi | tile_dim3 = 0 (unused)

    i32x4 g3 = {};
    g3[1] = 0x00010000;      // tensor_dim4 = 1 (unused), tile_dim4 = 0

    i32x8 gx2 = {};          // clang-23 6-arg form: extension operand, zero-filled

    __builtin_amdgcn_tensor_load_to_lds(g0, g1, g2, g3, gx2, 0);
    __builtin_amdgcn_s_wait_tensorcnt(0);
    asm volatile("" ::: "memory");
  }
  __syncthreads();

  const int ci = threadIdx.x & 31;   // channel within block == lane
  const int slot = threadIdx.x >> 5; // 8 waves x 32 pixels each
  float w[25];
#pragma unroll
  for (int k = 0; k < 25; ++k) w[k] = conv_w[(c0 + ci) * 25 + k];
  const float gv = gate[b * CC + c0 + ci];

  for (int q = 0; q < 32; ++q) {
    const int pix = slot * 32 + q;
    const int ly = pix >> 4;
    const int lx = pix & 15;
    float acc = 0.f;
#pragma unroll
    for (int ky = 0; ky < 5; ++ky) {
      const int hy = ly + ky;
      const bool yok = (hy >= oy) && (hy < oy + ny);
      const int ry = hy - oy;
#pragma unroll
      for (int kx = 0; kx < 5; ++kx) {
        const int hx = lx + kx;
        const bool ok = yok && (hx >= ox) && (hx < ox + nx);
        const int idx = ok ? ((ry * nx + (hx - ox)) * 32 + ci) : 0;
        const float v = ok ? tile[idx] : 0.f;
        acc += w[ky * 5 + kx] * v;  // lanes read 32 consecutive floats: no bank conflicts
      }
    }
    const int y = gy * TILE + ly;
    const int x = gx * TILE + lx;
    out[((size_t)(y * WW + x) * BS + b) * CC + c0 + ci] = gv * acc;  // 128B/wave store
  }
}

// ---------------------------------------------------------------------------
extern "C" void kernel_launch(void* const* d_in, const int* in_sizes, int n_in,
                              void* d_out, int out_size, void* d_ws, size_t ws_size,
                              hipStream_t stream) {
  const float* x = (const float*)d_in[0];
  const float* gn_w = (const float*)d_in[1];
  const float* gn_b = (const float*)d_in[2];
  const float* alpha = (const float*)d_in[3];
  const float* gamma = (const float*)d_in[4];
  const float* beta = (const float*)d_in[5];
  const float* conv_w = (const float*)d_in[6];
  // d_in[7], d_in[8] = h, w (fixed 128x128)
  float* out = (float*)d_out;

  // Workspace layout (floats): xa | part1 | meanrstd | part2 | gate
  float* xa = (float*)d_ws;                                  // 33554432
  float* part1 = xa + (size_t)HWPIX * BS * CC;               // + 32768
  float* meanrstd = part1 + (size_t)BS * PCHUNKS * 32 * 2;   // + 512
  float* part2 = meanrstd + (size_t)BS * GROUPS * 2;         // + 131072
  float* gate = part2 + (size_t)BS * PCHUNKS * CC;           // + 2048
  // total ~128.7 MiB of d_ws; every cell is written before read each call.

  k_stats_partial<<<dim3(PCHUNKS, BS), 256, 0, stream>>>(x, part1);
  k_stats_final<<<1, 256, 0, stream>>>(part1, meanrstd);
  k_norm_gelu<<<dim3(PCHUNKS, BS), 256, 0, stream>>>(x, gn_w, gn_b, meanrstd, xa, part2);
  k_gate<<<BS, 256, 0, stream>>>(part2, alpha, gamma, beta, gate);
  k_dwconv<<<dim3(CC / 32, 64, BS), 256, 0, stream>>>(xa, conv_w, gate, out);
}